// Head_73744588472619
// MI455X (gfx1250) — compile-verified
//
#include <hip/hip_runtime.h>
#include <hip/hip_bf16.h>
#include <stdint.h>

// Shapes from the reference: x[B,S,C], W*[C,H] -> out[B,S,H]
#define Bsz 64
#define Ssz 1024
#define Csz 64
#define Hsz 16
#define BSH (Bsz * Ssz * Hsz)   // elements per projected matrix (q/k/v)

typedef __attribute__((ext_vector_type(2))) float v2f;
typedef __attribute__((ext_vector_type(8))) float v8f;
typedef __attribute__((ext_vector_type(4))) int   i4;

// ---------------------------------------------------------------------------
// CDNA5 async global->LDS copy (ASYNCcnt-tracked), guarded so the file still
// compiles if this toolchain doesn't declare the builtins.
// ---------------------------------------------------------------------------
#define HAVE_ASYNC_LDS 0
#define HAVE_WAIT_ASYNC 0
#if defined(__has_builtin)
#if __has_builtin(__builtin_amdgcn_global_load_async_to_lds_b128)
#undef HAVE_ASYNC_LDS
#define HAVE_ASYNC_LDS 1
#endif
#if __has_builtin(__builtin_amdgcn_s_wait_asynccnt)
#undef HAVE_WAIT_ASYNC
#define HAVE_WAIT_ASYNC 1
#endif
#endif

typedef __attribute__((address_space(1))) i4 gi4;   // global 16B vector
typedef __attribute__((address_space(3))) i4 li4;   // LDS 16B vector

__device__ __forceinline__ void async_copy_b128(const float* gsrc, float* ldst) {
#if HAVE_ASYNC_LDS
    // Flat pointers to __shared__ carry the LDS byte offset in the low 32
    // bits (ISA aperture rule), so truncation yields the AS3 pointer value.
    __builtin_amdgcn_global_load_async_to_lds_b128(
        (gi4*)(uintptr_t)gsrc,
        (li4*)(uint32_t)(uintptr_t)ldst,
        0, 0);
#else
    float4 t = *(const float4*)gsrc;   // synchronous fallback
    *(float4*)ldst = t;
#endif
}

__device__ __forceinline__ void wait_async0() {
#if HAVE_WAIT_ASYNC
    __builtin_amdgcn_s_wait_asynccnt(0);
#elif HAVE_ASYNC_LDS
    asm volatile("s_wait_asynccnt 0x0" ::: "memory");
#endif
}

// ---------------------------------------------------------------------------
// Kernel 1: fused Q/K/V projection — 48 f32 WMMAs per wave, streams x once,
// writes q/k/v scratch.
// ---------------------------------------------------------------------------
__global__ __launch_bounds__(128) void
proj_qkv_kernel(const float* __restrict__ x,
                const float* __restrict__ Wq,
                const float* __restrict__ Wk,
                const float* __restrict__ Wv,
                float* __restrict__ q,
                float* __restrict__ k,
                float* __restrict__ v)
{
    const int lane = threadIdx.x & 31;
    const int wave = threadIdx.x >> 5;
    const int tile = blockIdx.x * 4 + wave;       // 16-row tile, 0..4095
    const int row0 = tile * 16;
    const int m    = lane & 15;
    const int half = lane >> 4;

    const float* xrow = x + (size_t)(row0 + m) * Csz;

    v8f cq = {}; v8f ck = {}; v8f cv = {};
    #pragma unroll
    for (int kk = 0; kk < 16; ++kk) {
        const int c0 = 4 * kk + 2 * half;
        v2f a = *(const v2f*)(xrow + c0);
        v2f bq, bk, bv;
        bq[0] = Wq[(c0    ) * Hsz + m];
        bq[1] = Wq[(c0 + 1) * Hsz + m];
        bk[0] = Wk[(c0    ) * Hsz + m];
        bk[1] = Wk[(c0 + 1) * Hsz + m];
        bv[0] = Wv[(c0    ) * Hsz + m];
        bv[1] = Wv[(c0 + 1) * Hsz + m];
        cq = __builtin_amdgcn_wmma_f32_16x16x4_f32(false, a, false, bq, (short)0, cq, false, false);
        ck = __builtin_amdgcn_wmma_f32_16x16x4_f32(false, a, false, bk, (short)0, ck, false, false);
        cv = __builtin_amdgcn_wmma_f32_16x16x4_f32(false, a, false, bv, (short)0, cv, false, false);
    }

    #pragma unroll
    for (int r = 0; r < 8; ++r) {
        const size_t off = (size_t)(row0 + r + 8 * half) * Hsz + m;
        q[off] = cq[r];
        k[off] = ck[r];
        v[off] = cv[r];
    }
}

// ---------------------------------------------------------------------------
// Kernel 2: flash-style attention.  A block (4 waves) owns 4 query tiles of
// ONE batch and sweeps the 64 key tiles.  Each 2 KB K/V tile is staged into
// LDS with async b128 copies (one per thread), double-buffered so the
// prefetch of tile j+1 overlaps the 8 WMMAs + softmax VALU of tile j.
// ---------------------------------------------------------------------------
__global__ __launch_bounds__(128) void
attn_kernel(const float* __restrict__ q,
            const float* __restrict__ k,
            const float* __restrict__ v,
            float* __restrict__ out)
{
    __shared__ float kvbuf[2][2][256];   // [buf][0=K,1=V][16x16]  (4 KB)
    __shared__ float plds[4][256];       // per-wave P tile patch  (4 KB)

    const int t    = threadIdx.x;
    const int lane = t & 31;
    const int wave = t >> 5;
    const int b    = blockIdx.x >> 4;              // batch
    const int qt   = (blockIdx.x & 15) * 4 + wave; // query tile within batch
    const int m    = lane & 15;
    const int half = lane >> 4;

    const float* qbase = q + ((size_t)b * Ssz + (size_t)qt * 16) * Hsz;
    const float* kbase = k + (size_t)b * Ssz * Hsz;
    const float* vbase = v + (size_t)b * Ssz * Hsz;

    // cooperative stage of key/value tile j into kvbuf[buf]
    const int sel = t >> 6;        // 0: K-half of the block, 1: V-half
    const int idx = t & 63;        // which 16-byte chunk of the 1 KB tile
    const float* stage_src0 = (sel ? vbase : kbase) + idx * 4;
    float*       stage_dst0 = &kvbuf[0][sel][idx * 4];
    float*       stage_dst1 = &kvbuf[1][sel][idx * 4];

    // Q tile in A-operand layout, softmax scale 64^-0.5 folded in.
    const float scale = 0.125f;
    v2f aq[4];
    #pragma unroll
    for (int kk = 0; kk < 4; ++kk) {
        v2f tq = *(const v2f*)(qbase + m * Hsz + 4 * kk + 2 * half);
        aq[kk][0] = tq[0] * scale;
        aq[kk][1] = tq[1] * scale;
    }

    float mrow[8], lrow[8];
    #pragma unroll
    for (int r = 0; r < 8; ++r) { mrow[r] = -1e30f; lrow[r] = 0.0f; }
    v8f acc = {};

    float* myP = &plds[wave][0];

    // preload tile 0 into buffer 0
    async_copy_b128(stage_src0, stage_dst0);
    wait_async0();
    __syncthreads();

    for (int j = 0; j < 64; ++j) {
        // prefetch tile j+1 into the other buffer (overlaps compute below)
        if (j < 63) {
            async_copy_b128(stage_src0 + (size_t)(j + 1) * 256,
                            ((j + 1) & 1) ? stage_dst1 : stage_dst0);
        }

        const float* kK = &kvbuf[j & 1][0][0];
        const float* kV = &kvbuf[j & 1][1][0];

        // S = (Q*scale) x K^T : 4 chained f32 WMMAs (K-dim = H = 16)
        v8f s = {};
        #pragma unroll
        for (int kk = 0; kk < 4; ++kk) {
            v2f bk = *(const v2f*)(kK + m * 16 + 4 * kk + 2 * half);
            s = __builtin_amdgcn_wmma_f32_16x16x4_f32(false, aq[kk], false, bk, (short)0, s, false, false);
        }

        // Online softmax update (rows live in 16-lane groups; xor masks < 16
        // never cross the wave32 half boundary).
        #pragma unroll
        for (int r = 0; r < 8; ++r) {
            float val  = s[r];
            float rmax = val;
            rmax = fmaxf(rmax, __shfl_xor(rmax, 1));
            rmax = fmaxf(rmax, __shfl_xor(rmax, 2));
            rmax = fmaxf(rmax, __shfl_xor(rmax, 4));
            rmax = fmaxf(rmax, __shfl_xor(rmax, 8));
            const float mnew = fmaxf(mrow[r], rmax);
            const float p    = __expf(val - mnew);
            float rsum = p;
            rsum += __shfl_xor(rsum, 1);
            rsum += __shfl_xor(rsum, 2);
            rsum += __shfl_xor(rsum, 4);
            rsum += __shfl_xor(rsum, 8);
            const float alpha = __expf(mrow[r] - mnew);
            lrow[r] = lrow[r] * alpha + rsum;
            mrow[r] = mnew;
            acc[r] *= alpha;
            myP[(r + 8 * half) * 16 + m] = p;   // stage P row-major in LDS
        }

        // acc += P x V_j : 4 chained f32 WMMAs.  P re-read from this wave's
        // private LDS patch in A-operand layout (same-wave LDS is in-order);
        // V operand read from the shared staged tile.
        #pragma unroll
        for (int kk = 0; kk < 4; ++kk) {
            v2f ap = *(const v2f*)(myP + m * 16 + 4 * kk + 2 * half);
            const int k0 = 4 * kk + 2 * half;
            v2f bv;
            bv[0] = kV[(k0    ) * 16 + m];
            bv[1] = kV[(k0 + 1) * 16 + m];
            acc = __builtin_amdgcn_wmma_f32_16x16x4_f32(false, ap, false, bv, (short)0, acc, false, false);
        }

        // make tile j+1 visible to all waves; also fences re-use of buf j
        wait_async0();
        __syncthreads();
    }

    // Normalize by the row sums and store.
    #pragma unroll
    for (int r = 0; r < 8; ++r) {
        const size_t off = ((size_t)b * Ssz + (size_t)qt * 16 + r + 8 * half) * Hsz + m;
        out[off] = acc[r] / lrow[r];
    }
}

extern "C" void kernel_launch(void* const* d_in, const int* in_sizes, int n_in,
                              void* d_out, int out_size, void* d_ws, size_t ws_size,
                              hipStream_t stream) {
    const float* x  = (const float*)d_in[0];
    const float* Wq = (const float*)d_in[1];
    const float* Wk = (const float*)d_in[2];
    const float* Wv = (const float*)d_in[3];

    float* ws  = (float*)d_ws;
    float* qws = ws;
    float* kws = ws + BSH;
    float* vws = ws + 2 * BSH;      // needs 3*B*S*H*4 = 12.6 MB of d_ws

    // Projection: 4096 sixteen-row tiles, 4 waves per block.
    proj_qkv_kernel<<<1024, 128, 0, stream>>>(x, Wq, Wk, Wv, qws, kws, vws);
    // Attention: 64 batches x 16 blocks, each block = 4 query tiles.
    attn_kernel<<<1024, 128, 0, stream>>>(qws, kws, vws, (float*)d_out);
}